// BiDecoder_22497038697227
// MI455X (gfx1250) — compile-verified
//
#include <hip/hip_runtime.h>
#include <hip/hip_bf16.h>

#ifndef D_DIM
#define D_DIM 128
#endif
#define NB 2
#define NC 5

typedef __attribute__((ext_vector_type(2))) float v2f;
typedef __attribute__((ext_vector_type(8))) float v8f;

// -----------------------------------------------------------------------------
// Kernel 1: hu[s] = ufeat @ Ps[s]   (NUxD) x (DxD), f32 WMMA 16x16x4
// grid = (ceil(NU/128), NB), block = 256 (8 waves). Each wave owns a 16-row
// slab and 8 N-tiles of 16 cols.
//
// Ps[s] is staged in LDS pair-interleaved: element (k, c) lives at
//   Plds[(k>>1)*PSTRIDE + c*2 + (k&1)]
// so a lane's B fragment (rows k+koff, k+koff+1 at column c) is one aligned
// 8-byte ds_load_b64. PSTRIDE=288 dwords => pair-row stride ≡ 32 (mod 64
// banks), making low/high half-wave reads bank-disjoint.
// -----------------------------------------------------------------------------
__global__ __launch_bounds__(256) void bidecoder_gemm_kernel(
    const float* __restrict__ ufeat,
    const float* __restrict__ Ps,
    float* __restrict__ hu,
    int NU)
{
    constexpr int PSTRIDE = 288;                    // dwords per K-pair row
    __shared__ float Plds[(D_DIM / 2) * PSTRIDE];   // 64*288*4 = 72 KB

    const int s = blockIdx.y;                       // basis index
    const float* __restrict__ Pg = Ps + (size_t)s * D_DIM * D_DIM;

    // Stage Ps[s] into pair-interleaved LDS. Global reads fully coalesced.
    for (int i = threadIdx.x; i < D_DIM * D_DIM; i += 256) {
        int r = i >> 7;               // k row
        int c = i & 127;              // column
        Plds[(r >> 1) * PSTRIDE + c * 2 + (r & 1)] = Pg[i];
    }
    __syncthreads();

    const int lane = threadIdx.x & 31;
    const int wave = threadIdx.x >> 5;
    const int ml   = lane & 15;                     // lane within half-wave
    const int hi   = lane >> 4;                     // 0 = low half, 1 = high half
    const int koff = hi << 1;                       // K sub-offset: 0 or 2

    const int m0   = blockIdx.x * 128 + wave * 16;  // first row of this wave's tile
    // Row clamp: an A-row >= NU only feeds a C-row >= NU, which is never
    // stored, so clamping (instead of predicating) is safe and branch-free.
    const int rowA = min(m0 + ml, NU - 1);
    const float* __restrict__ Arow = ufeat + (size_t)rowA * D_DIM;

    v8f acc[8] = {};                                // 8 N-tiles of 16x16 f32

    for (int k = 0; k < D_DIM; k += 4) {
        // A fragment: K = k+koff, k+koff+1 are contiguous -> one b64 load.
        v2f a = *(const v2f*)(Arow + k + koff);

        // B fragments: pair-row for K = k+koff (k+koff is even).
        const float* __restrict__ Bp =
            &Plds[((k + koff) >> 1) * PSTRIDE + ml * 2];
#pragma unroll
        for (int t = 0; t < 8; ++t) {
            v2f b = *(const v2f*)(Bp + t * 32);     // col = t*16 + ml
            acc[t] = __builtin_amdgcn_wmma_f32_16x16x4_f32(
                false, a, false, b, (short)0, acc[t], false, false);
        }
    }

    // Store C: VGPR r -> row m0 + r + 8*hi, col t*16 + ml
    float* __restrict__ C = hu + (size_t)s * NU * D_DIM;
    const int rbase = m0 + (hi << 3);
    if (m0 + 16 <= NU) {
        // Wave-uniform fast path: no predication.
#pragma unroll
        for (int t = 0; t < 8; ++t) {
#pragma unroll
            for (int r = 0; r < 8; ++r) {
                C[(size_t)(rbase + r) * D_DIM + t * 16 + ml] = acc[t][r];
            }
        }
    } else {
#pragma unroll
        for (int t = 0; t < 8; ++t) {
#pragma unroll
            for (int r = 0; r < 8; ++r) {
                int rr = rbase + r;
                if (rr < NU) C[(size_t)rr * D_DIM + t * 16 + ml] = acc[t][r];
            }
        }
    }
}

// -----------------------------------------------------------------------------
// Kernel 2: per-edge gather + dual dot + 5-class combine (bandwidth bound,
// hu + ifeat are L2-resident: 154 MB < 192 MB L2).
// -----------------------------------------------------------------------------
__global__ __launch_bounds__(256) void bidecoder_edge_kernel(
    const float* __restrict__ hu,     // [NB][NU][D]
    const float* __restrict__ ifeat,  // [NI][D]
    const float* __restrict__ Wc,     // [NC][NB]
    const int*   __restrict__ src,
    const int*   __restrict__ dst,
    float* __restrict__ out,          // [E][NC]
    int E, int NU)
{
    __shared__ float Wsh[NC * NB];
    if (threadIdx.x < NC * NB) Wsh[threadIdx.x] = Wc[threadIdx.x];
    __syncthreads();

    const int e = blockIdx.x * blockDim.x + threadIdx.x;
    if (e >= E) return;

    const int su = src[e];
    const int di = dst[e];

    const float4* __restrict__ u0 = (const float4*)(hu + (size_t)su * D_DIM);
    const float4* __restrict__ u1 = (const float4*)(hu + (size_t)NU * D_DIM + (size_t)su * D_DIM);
    const float4* __restrict__ vv = (const float4*)(ifeat + (size_t)di * D_DIM);

    float acc0 = 0.0f, acc1 = 0.0f;
#pragma unroll 8
    for (int i = 0; i < D_DIM / 4; ++i) {
        float4 v = vv[i];
        float4 a = u0[i];
        float4 b = u1[i];
        acc0 = fmaf(a.x, v.x, fmaf(a.y, v.y, fmaf(a.z, v.z, fmaf(a.w, v.w, acc0))));
        acc1 = fmaf(b.x, v.x, fmaf(b.y, v.y, fmaf(b.z, v.z, fmaf(b.w, v.w, acc1))));
    }

    float* __restrict__ o = out + (size_t)e * NC;
#pragma unroll
    for (int c = 0; c < NC; ++c)
        o[c] = fmaf(acc0, Wsh[c * NB + 0], acc1 * Wsh[c * NB + 1]);
}

// -----------------------------------------------------------------------------
// Launch: inputs in setup_inputs() order:
//   0 ufeat (NU*D f32), 1 ifeat (NI*D f32), 2 Ps (NB*D*D f32),
//   3 W_combine (NC*NB f32), 4 src (E int), 5 dst (E int)
// d_ws holds hu: NB*NU*D f32 = 102.4 MB.
// -----------------------------------------------------------------------------
extern "C" void kernel_launch(void* const* d_in, const int* in_sizes, int n_in,
                              void* d_out, int out_size, void* d_ws, size_t ws_size,
                              hipStream_t stream) {
    const float* ufeat = (const float*)d_in[0];
    const float* ifeat = (const float*)d_in[1];
    const float* Ps    = (const float*)d_in[2];
    const float* Wc    = (const float*)d_in[3];
    const int*   src   = (const int*)d_in[4];
    const int*   dst   = (const int*)d_in[5];
    float*       out   = (float*)d_out;
    float*       hu    = (float*)d_ws;

    const int NU = in_sizes[0] / D_DIM;
    const int E  = in_sizes[4];

    dim3 g1((NU + 127) / 128, NB);
    bidecoder_gemm_kernel<<<g1, 256, 0, stream>>>(ufeat, Ps, hu, NU);

    int blocks = (E + 255) / 256;
    bidecoder_edge_kernel<<<blocks, 256, 0, stream>>>(hu, ifeat, Wc, src, dst, out, E, NU);
}